// Model_61005715472614
// MI455X (gfx1250) — compile-verified
//
#include <hip/hip_runtime.h>
#include <hip/hip_bf16.h>
#include <math.h>

#define NB 1024   // batch
#define NT 512    // time
#define NF 32     // input features
#define NH 64     // hidden
#define NG 256    // 4*NH gate width

typedef __attribute__((ext_vector_type(2))) float v2f;
typedef __attribute__((ext_vector_type(8))) float v8f;

#if defined(__has_builtin)
#if __has_builtin(__builtin_amdgcn_wmma_f32_16x16x4_f32)
#define HAVE_WMMA_F32 1
#endif
#endif

// D(16x16) = A(16x4) * B(4x16) + C, fp32 WMMA.
// A frag: lane L holds A[L&15][2*(L>>4) + {0,1}]
// B frag: lane L holds B[2*(L>>4) + {0,1}][L&15]
// C/D:    lane L, vgpr r holds D[r + 8*(L>>4)][L&15]
__device__ __forceinline__ v8f wmma4(v2f a, v2f b, v8f c) {
#ifdef HAVE_WMMA_F32
  return __builtin_amdgcn_wmma_f32_16x16x4_f32(false, a, false, b, (short)0, c,
                                               false, false);
#else
  // correct lane-shuffle emulation (compile-safe fallback)
  const int lane = threadIdx.x & 31;
  const int n = lane & 15;
  #pragma unroll
  for (int r = 0; r < 8; ++r) {
    const int m = r + 8 * (lane >> 4);
    float s = c[r];
    #pragma unroll
    for (int k = 0; k < 4; ++k) {
      float av = __shfl((k & 1) ? a.y : a.x, m + 16 * (k >> 1), 32);
      float bv = __shfl((k & 1) ? b.y : b.x, n + 16 * (k >> 1), 32);
      s += av * bv;
    }
    c[r] = s;
  }
  return c;
#endif
}

// Workgroup barrier that drains ONLY the LDS counter. The per-step h exchange
// lives in LDS; the BN'd h global store and the t+1 xz prefetch must NOT be
// drained here (that would put an L2/HBM round-trip on the serial recurrence
// path). Their ordering is handled by kernel boundaries / use-site waits.
__device__ __forceinline__ void lds_only_barrier() {
  asm volatile(
      "s_wait_dscnt 0x0\n\t"
      "s_barrier_signal -1\n\t"
      "s_barrier_wait -1" ::: "memory");
}

// ---------------------------------------------------------------------------
// xz[m, 0:256] = in[m, 0:FIN] @ Wx[FIN, 256] + bias, m = b*T + t
// 128 threads (4 waves); block handles one 16-row M tile, wave w -> N tiles 4w..4w+3
// ---------------------------------------------------------------------------
template <int FIN>
__global__ __launch_bounds__(128) void proj_kernel(
    const float* __restrict__ in, const float* __restrict__ Wx,
    const float* __restrict__ bias, float* __restrict__ xz) {
  constexpr int STR = FIN + 2;  // even pad -> conflict-free column reads
  __shared__ float xs[16 * STR];
  const int tid = threadIdx.x, lane = tid & 31, wave = tid >> 5;
  const long m0 = (long)blockIdx.x * 16;

  for (int idx = tid; idx < 16 * FIN; idx += 128) {
    int m = idx / FIN, k = idx % FIN;
    xs[m * STR + k] = in[(m0 + m) * FIN + k];
  }
  __syncthreads();

  const int n = lane & 15;
  const int kb = (lane >> 4) * 2;

  v2f afr[FIN / 4];
  #pragma unroll
  for (int kk = 0; kk < FIN / 4; ++kk) {
    afr[kk].x = xs[(lane & 15) * STR + kk * 4 + kb];
    afr[kk].y = xs[(lane & 15) * STR + kk * 4 + kb + 1];
  }

  #pragma unroll
  for (int nt = 0; nt < 4; ++nt) {
    const int n0 = (wave * 4 + nt) * 16;
    const float bv = bias[n0 + n];
    v8f acc;
    #pragma unroll
    for (int r = 0; r < 8; ++r) acc[r] = bv;
    #pragma unroll
    for (int kk = 0; kk < FIN / 4; ++kk) {
      v2f bfr;
      bfr.x = Wx[(kk * 4 + kb) * NG + n0 + n];
      bfr.y = Wx[(kk * 4 + kb + 1) * NG + n0 + n];
      acc = wmma4(afr[kk], bfr, acc);
    }
    #pragma unroll
    for (int r = 0; r < 8; ++r) {
      const int m = r + 8 * (lane >> 4);
      xz[(m0 + m) * NG + n0 + n] = acc[r];
    }
  }
}

// ---------------------------------------------------------------------------
// Persistent LSTM recurrence over all T steps for a 16-row batch tile.
// 4 waves; wave w owns hidden cols [16w,16w+16) and its 4 gate tiles (i,f,g,o).
// Uh fragments live in VGPRs; h double-buffered in LDS; BN fused into h store.
// ---------------------------------------------------------------------------
__global__ __launch_bounds__(128) void lstm_kernel(
    const float* __restrict__ xz, const float* __restrict__ Uh,
    const float* __restrict__ gamma, const float* __restrict__ beta,
    const float* __restrict__ mean, const float* __restrict__ var,
    float* __restrict__ hout) {
  __shared__ float hbuf[2][16 * 66];  // stride 66: conflict-free, b64-aligned
  const int tid = threadIdx.x, lane = tid & 31, wave = tid >> 5;
  const int n = lane & 15;
  const int kb = (lane >> 4) * 2;
  const int j = wave * 16 + n;  // hidden column this lane produces
  const long b0 = (long)blockIdx.x * 16;

  const float bnsc = gamma[j] * rsqrtf(var[j] + 1e-3f);
  const float bnbs = beta[j] - mean[j] * bnsc;

  // Preload recurrent weight fragments: ub[gate][kstep]
  v2f ub[4][16];
  #pragma unroll
  for (int g = 0; g < 4; ++g) {
    const int n0 = g * NH + wave * 16;
    #pragma unroll
    for (int kk = 0; kk < 16; ++kk) {
      ub[g][kk].x = Uh[(kk * 4 + kb) * NG + n0 + n];
      ub[g][kk].y = Uh[(kk * 4 + kb + 1) * NG + n0 + n];
    }
  }

  v8f cst;
  #pragma unroll
  for (int r = 0; r < 8; ++r) cst[r] = 0.0f;

  // prefetch xz for t=0
  v8f zn[4];
  #pragma unroll
  for (int g = 0; g < 4; ++g) {
    const int n0 = g * NH + wave * 16;
    #pragma unroll
    for (int r = 0; r < 8; ++r) {
      const int m = r + 8 * (lane >> 4);
      zn[g][r] = xz[((b0 + m) * NT + 0) * NG + n0 + n];
    }
  }

  for (int t = 0; t < NT; ++t) {
    v8f acc[4];
    #pragma unroll
    for (int g = 0; g < 4; ++g) acc[g] = zn[g];

    if (t + 1 < NT) {  // prefetch next step's xz; overlaps the WMMA chain
      #pragma unroll
      for (int g = 0; g < 4; ++g) {
        const int n0 = g * NH + wave * 16;
        #pragma unroll
        for (int r = 0; r < 8; ++r) {
          const int m = r + 8 * (lane >> 4);
          zn[g][r] = xz[((b0 + m) * NT + (t + 1)) * NG + n0 + n];
        }
      }
    }

    if (t > 0) {  // h_0 == 0: skip recurrent GEMM on first step
      const float* hb = hbuf[t & 1];
      #pragma unroll
      for (int kk = 0; kk < 16; ++kk) {
        v2f a;
        a.x = hb[(lane & 15) * 66 + kk * 4 + kb];
        a.y = hb[(lane & 15) * 66 + kk * 4 + kb + 1];
        #pragma unroll
        for (int g = 0; g < 4; ++g) acc[g] = wmma4(a, ub[g][kk], acc[g]);
      }
    }

    float* hn = hbuf[(t + 1) & 1];
    #pragma unroll
    for (int r = 0; r < 8; ++r) {
      const float iv = 1.0f / (1.0f + __expf(-acc[0][r]));
      const float fv = 1.0f / (1.0f + __expf(-acc[1][r]));
      const float gv = fmaxf(acc[2][r], 0.0f);
      const float ov = 1.0f / (1.0f + __expf(-acc[3][r]));
      const float cn = fv * cst[r] + iv * gv;
      cst[r] = cn;
      const float hv = ov * fmaxf(cn, 0.0f);
      const int m = r + 8 * (lane >> 4);
      hn[m * 66 + j] = hv;                                    // raw h for recurrence
      hout[((b0 + m) * NT + t) * NH + j] = hv * bnsc + bnbs;  // fused BN
    }
    // LDS-only split barrier: keeps the hout store + xz prefetch off the
    // serial per-step critical path.
    lds_only_barrier();
  }
}

// ---------------------------------------------------------------------------
// attention: wv = attw@attv; score[t]=h[b,t,:]·wv; softmax over t; ctx = h^T@alpha
// one block per batch row
// ---------------------------------------------------------------------------
__global__ __launch_bounds__(256) void attn_kernel(
    const float* __restrict__ h, const float* __restrict__ attw,
    const float* __restrict__ attv, float* __restrict__ feat, int layer) {
  __shared__ float wv[NH];
  __shared__ float sc[NT];
  __shared__ float red[256];
  __shared__ float part[256];
  const int tid = threadIdx.x;
  const int b = blockIdx.x;

  if (tid < NH) {
    float s = 0.0f;
    #pragma unroll 8
    for (int e = 0; e < NH; ++e) s += attw[tid * NH + e] * attv[e];
    wv[tid] = s;
  }
  __syncthreads();

  // pass 1: scores for t = tid, tid+256
  float s0 = 0.0f, s1 = 0.0f;
  {
    const float* hp0 = h + ((size_t)b * NT + tid) * NH;
    const float* hp1 = h + ((size_t)b * NT + tid + 256) * NH;
    #pragma unroll 16
    for (int jj = 0; jj < NH; ++jj) {
      s0 += hp0[jj] * wv[jj];
      s1 += hp1[jj] * wv[jj];
    }
  }

  // max reduce
  red[tid] = fmaxf(s0, s1);
  __syncthreads();
  for (int off = 128; off > 0; off >>= 1) {
    if (tid < off) red[tid] = fmaxf(red[tid], red[tid + off]);
    __syncthreads();
  }
  const float mx = red[0];
  __syncthreads();

  const float e0 = __expf(s0 - mx);
  const float e1 = __expf(s1 - mx);
  sc[tid] = e0;
  sc[tid + 256] = e1;
  red[tid] = e0 + e1;
  __syncthreads();
  for (int off = 128; off > 0; off >>= 1) {
    if (tid < off) red[tid] += red[tid + off];
    __syncthreads();
  }
  const float inv = 1.0f / red[0];
  __syncthreads();

  // pass 2: weighted sum over time, coalesced on hidden dim
  const int jl = tid & 63;
  const int slc = tid >> 6;  // 4 time slices of 128
  float acc = 0.0f;
  for (int tt = 0; tt < 128; ++tt) {
    const int t = slc * 128 + tt;
    acc += sc[t] * h[((size_t)b * NT + t) * NH + jl];
  }
  part[tid] = acc;
  __syncthreads();
  if (tid < NH) {
    feat[(size_t)b * (3 * NH) + layer * NH + tid] =
        (part[tid] + part[tid + 64] + part[tid + 128] + part[tid + 192]) * inv;
  }
}

// out[b] = feat[b,:]·dw + db
__global__ __launch_bounds__(256) void head_kernel(
    const float* __restrict__ feat, const float* __restrict__ dw,
    const float* __restrict__ db, float* __restrict__ out) {
  const int b = blockIdx.x * blockDim.x + threadIdx.x;
  if (b < NB) {
    float s = db[0];
    #pragma unroll 8
    for (int k = 0; k < 3 * NH; ++k) s += feat[(size_t)b * (3 * NH) + k] * dw[k];
    out[b] = s;
  }
}

extern "C" void kernel_launch(void* const* d_in, const int* in_sizes, int n_in,
                              void* d_out, int out_size, void* d_ws,
                              size_t ws_size, hipStream_t stream) {
  (void)in_sizes; (void)n_in; (void)out_size; (void)ws_size;
  const float* x = (const float*)d_in[0];
  // per layer l (0..2), base = 1 + 9*l: w,u,bi,gamma,beta,mean,var,attw,attv
  auto P = [&](int l, int k) { return (const float*)d_in[1 + 9 * l + k]; };
  const float* dw = (const float*)d_in[28];
  const float* db = (const float*)d_in[29];
  float* out = (float*)d_out;

  float* xz = (float*)d_ws;                       // [B*T, 256]
  float* hb = xz + (size_t)NB * NT * NG;          // [B*T, 64]
  float* feat = hb + (size_t)NB * NT * NH;        // [B, 192]

  const dim3 gProj(NB * NT / 16), bProj(128);
  const dim3 gLstm(NB / 16), bLstm(128);
  const dim3 gAttn(NB), bAttn(256);

  // ---- layer 1 ----
  proj_kernel<NF><<<gProj, bProj, 0, stream>>>(x, P(0, 0), P(0, 2), xz);
  lstm_kernel<<<gLstm, bLstm, 0, stream>>>(xz, P(0, 1), P(0, 3), P(0, 4),
                                           P(0, 5), P(0, 6), hb);
  attn_kernel<<<gAttn, bAttn, 0, stream>>>(hb, P(0, 7), P(0, 8), feat, 0);
  // ---- layer 2 ----
  proj_kernel<NH><<<gProj, bProj, 0, stream>>>(hb, P(1, 0), P(1, 2), xz);
  lstm_kernel<<<gLstm, bLstm, 0, stream>>>(xz, P(1, 1), P(1, 3), P(1, 4),
                                           P(1, 5), P(1, 6), hb);
  attn_kernel<<<gAttn, bAttn, 0, stream>>>(hb, P(1, 7), P(1, 8), feat, 1);
  // ---- layer 3 ----
  proj_kernel<NH><<<gProj, bProj, 0, stream>>>(hb, P(2, 0), P(2, 2), xz);
  lstm_kernel<<<gLstm, bLstm, 0, stream>>>(xz, P(2, 1), P(2, 3), P(2, 4),
                                           P(2, 5), P(2, 6), hb);
  attn_kernel<<<gAttn, bAttn, 0, stream>>>(hb, P(2, 7), P(2, 8), feat, 2);
  // ---- head ----
  head_kernel<<<dim3(NB / 256), dim3(256), 0, stream>>>(feat, dw, db, out);
}